// VectorQuantizerEMA_68375879352395
// MI455X (gfx1250) — compile-verified
//
#include <hip/hip_runtime.h>
#include <hip/hip_bf16.h>

typedef __attribute__((ext_vector_type(2))) float v2f;
typedef __attribute__((ext_vector_type(8))) float v8f;

#define NUM_EMB   1024
#define EMB_DIM   64
#define BATCH     64
#define HW        1024          // 32*32
#define NPTS      65536         // BATCH*HW
#define PITCH     68            // LDS row pitch (floats): conflict-free b64 reads
#define DECAYF    0.99f
#define COMMITF   0.25f
#define EPSF      1e-5f

// ---- output layout (floats, concatenated in reference return order) ----
#define OQ 0                    // z_q_st            4,194,304
#define OL 4194304              // loss              1
#define OI 4194305              // indices (as f32)  65,536
#define OC 4259841              // new_cluster_size  1,024
#define OW 4260865              // new_ema_w         65,536
#define OE 4326401              // new_embedding     65,536

// ---- workspace layout (floats) ----
#define WS_ENORM  0             // 1024
#define WS_COUNTS 1024          // 1024
#define WS_DW     2048          // 65536
#define WS_LOSS   67584         // 1
#define WS_ZERON  66561         // counts+dw+loss contiguous zero region

// -------------------------------------------------------------------------
__global__ void vq_zero_kernel(float* __restrict__ p, int n) {
    int i = blockIdx.x * blockDim.x + threadIdx.x;
    if (i < n) p[i] = 0.0f;
}

// -------------------------------------------------------------------------
// e_norm[k] = sum_d emb[k][d]^2
__global__ void vq_enorm_kernel(const float* __restrict__ emb,
                                float* __restrict__ enorm) {
    int k = blockIdx.x * blockDim.x + threadIdx.x;
    if (k < NUM_EMB) {
        const float4* row = reinterpret_cast<const float4*>(emb + k * EMB_DIM);
        float s = 0.0f;
#pragma unroll
        for (int j = 0; j < EMB_DIM / 4; ++j) {
            float4 v = row[j];
            s += v.x * v.x + v.y * v.y + v.z * v.z + v.w * v.w;
        }
        enorm[k] = s;
    }
}

// -------------------------------------------------------------------------
// Main kernel: per-wave 32-row M-tile, loop over 64 code tiles of 16.
// score = ||e||^2 - 2 * (z . e)  via chained v_wmma_f32_16x16x4_f32.
__global__ __launch_bounds__(256) void vq_argmin_kernel(
    const float* __restrict__ z,       // (B, D, H, W)
    const float* __restrict__ emb,     // (K, D)
    const float* __restrict__ enorm,   // (K,)
    float* __restrict__ out_idx_f)     // (N,) indices as float
{
    extern __shared__ float lds[];     // NUM_EMB * PITCH floats (272 KB)
    const int tid = threadIdx.x;

    // Stage full codebook into LDS with pitch-68 rows (16B-aligned rows).
    for (int j = tid; j < NUM_EMB * EMB_DIM / 4; j += 256) {
        float4 v = reinterpret_cast<const float4*>(emb)[j];
        int c = j >> 4;                // j*4 / 64
        int d = (j & 15) << 2;
        *reinterpret_cast<float4*>(&lds[c * PITCH + d]) = v;
    }
    __syncthreads();

    const int wave = tid >> 5;
    const int lane = tid & 31;
    const int lrow = lane & 15;        // M (for A) / N (for B) within tile
    const int lhi  = lane >> 4;        // lane-half selects K sub-pair

    const int n0 = blockIdx.x * 256 + wave * 32;   // 32 rows per wave
    const int b  = n0 >> 10;                       // whole WG inside one batch
    const float* zb = z + (size_t)b * (EMB_DIM * HW);
    const int p0 = n0 & (HW - 1);

    // A fragments for both 16-row subtiles, all of D=64 (ISA 16x4 fp32 layout):
    // lane: M = lrow;  VGPR0 = K(dbase), VGPR1 = K(dbase+1), dbase = 4*kk + 2*lhi
    float ax[2][16], ay[2][16];
#pragma unroll
    for (int m = 0; m < 2; ++m) {
        int p = p0 + m * 16 + lrow;
#pragma unroll
        for (int kk = 0; kk < 16; ++kk) {
            int dbase = kk * 4 + lhi * 2;
            ax[m][kk] = zb[dbase * HW + p];
            ay[m][kk] = zb[(dbase + 1) * HW + p];
        }
    }

    float bestv[2][8];
    int   besti[2][8];
#pragma unroll
    for (int m = 0; m < 2; ++m)
#pragma unroll
        for (int r = 0; r < 8; ++r) { bestv[m][r] = 3.4e38f; besti[m][r] = 0; }

#pragma unroll 1
    for (int ct = 0; ct < NUM_EMB / 16; ++ct) {
        const int   c  = ct * 16 + lrow;      // this lane's code column
        const float en = enorm[c];

        v8f acc0 = {};
        v8f acc1 = {};
#pragma unroll
        for (int kk = 0; kk < 16; ++kk) {
            int dbase = kk * 4 + lhi * 2;
            v2f bb;
            bb.x = lds[c * PITCH + dbase];
            bb.y = lds[c * PITCH + dbase + 1];
            v2f a0; a0.x = ax[0][kk]; a0.y = ay[0][kk];
            v2f a1; a1.x = ax[1][kk]; a1.y = ay[1][kk];
            acc0 = __builtin_amdgcn_wmma_f32_16x16x4_f32(
                false, a0, false, bb, (short)0, acc0, false, false);
            acc1 = __builtin_amdgcn_wmma_f32_16x16x4_f32(
                false, a1, false, bb, (short)0, acc1, false, false);
        }

        // C/D layout: VGPR r holds row M = r + 8*lhi, col N = lrow (code c).
#pragma unroll
        for (int r = 0; r < 8; ++r) {
            float s0 = en - 2.0f * acc0[r];
            float s1 = en - 2.0f * acc1[r];
            if (s0 < bestv[0][r]) { bestv[0][r] = s0; besti[0][r] = c; }
            if (s1 < bestv[1][r]) { bestv[1][r] = s1; besti[1][r] = c; }
        }
    }

    // Reduce argmin over the 16 columns held by this lane's half-group;
    // tie-break to lowest index (matches jnp.argmin).
#pragma unroll
    for (int m = 0; m < 2; ++m) {
#pragma unroll
        for (int r = 0; r < 8; ++r) {
            float v = bestv[m][r];
            int   i = besti[m][r];
#pragma unroll
            for (int off = 1; off < 16; off <<= 1) {
                float ov = __shfl_xor(v, off, 32);
                int   oi = __shfl_xor(i, off, 32);
                if (ov < v || (ov == v && oi < i)) { v = ov; i = oi; }
            }
            besti[m][r] = i;
        }
    }

    // Each half-group lane lrow<8 writes row M = lrow + 8*lhi of each subtile.
#pragma unroll
    for (int m = 0; m < 2; ++m) {
        int sel = besti[m][0];
#pragma unroll
        for (int r = 1; r < 8; ++r) sel = (lrow == r) ? besti[m][r] : sel;
        if (lrow < 8) {
            int row = n0 + m * 16 + lrow + 8 * lhi;
            out_idx_f[row] = (float)sel;
        }
    }
}

// -------------------------------------------------------------------------
// Gather z_q into (B,D,H,W), accumulate dw / counts / loss partials.
// gid -> d = gid>>16 (fixed per 65536 block), n = gid & 65535  => z and out
// accesses are fully coalesced along H*W.
__global__ __launch_bounds__(256) void vq_scatter_kernel(
    const float* __restrict__ z, const float* __restrict__ emb,
    const float* __restrict__ idx_f,
    float* __restrict__ out_q, float* __restrict__ counts,
    float* __restrict__ dw, float* __restrict__ loss_sum)
{
    int gid = blockIdx.x * blockDim.x + threadIdx.x;
    int d = gid >> 16;
    int n = gid & (NPTS - 1);
    int b = n >> 10;
    int p = n & (HW - 1);
    int k = (int)idx_f[n];

    float zq = emb[k * EMB_DIM + d];            // 256 KB, L2-resident gather
    size_t zoff = (size_t)b * (EMB_DIM * HW) + (size_t)d * HW + p;
    float zv = z[zoff];
    out_q[zoff] = zq;                           // z + sg(z_q - z) == z_q

    atomicAdd(&dw[k * EMB_DIM + d], zv);        // segment-sum in L2

    float diff = zv - zq;
    float sq = diff * diff;
#pragma unroll
    for (int off = 16; off >= 1; off >>= 1) sq += __shfl_xor(sq, off, 32);
    if ((threadIdx.x & 31) == 0) atomicAdd(loss_sum, sq);
    if (d == 0) atomicAdd(&counts[k], 1.0f);
}

// -------------------------------------------------------------------------
// EMA finalize: one workgroup, one thread per code.
__global__ __launch_bounds__(1024) void vq_finalize_kernel(
    const float* __restrict__ cluster_size, const float* __restrict__ ema_w,
    const float* __restrict__ counts, const float* __restrict__ dw,
    const float* __restrict__ loss_sum,
    float* __restrict__ out_ncs, float* __restrict__ out_nemaw,
    float* __restrict__ out_nemb, float* __restrict__ out_loss)
{
    __shared__ float red[33];
    const int k = threadIdx.x;

    float ncs = DECAYF * cluster_size[k] + (1.0f - DECAYF) * counts[k];
    out_ncs[k] = ncs;

#pragma unroll 1
    for (int d = 0; d < EMB_DIM; ++d)
        out_nemaw[k * EMB_DIM + d] =
            DECAYF * ema_w[k * EMB_DIM + d] + (1.0f - DECAYF) * dw[k * EMB_DIM + d];

    // n = sum(new_cluster_size) over 1024 codes (32 waves)
    float s = ncs;
#pragma unroll
    for (int off = 16; off >= 1; off >>= 1) s += __shfl_xor(s, off, 32);
    if ((k & 31) == 0) red[k >> 5] = s;
    __syncthreads();
    if (k < 32) {
        float t = red[k];
#pragma unroll
        for (int off = 16; off >= 1; off >>= 1) t += __shfl_xor(t, off, 32);
        if (k == 0) red[32] = t;
    }
    __syncthreads();
    float ntot = red[32];

    float cs  = (ncs + EPSF) / (ntot + (float)NUM_EMB * EPSF) * ntot;
    float inv = 1.0f / cs;
#pragma unroll 1
    for (int d = 0; d < EMB_DIM; ++d)                       // same-thread re-read
        out_nemb[k * EMB_DIM + d] = out_nemaw[k * EMB_DIM + d] * inv;

    if (k == 0)
        out_loss[0] = COMMITF * loss_sum[0] / (float)(BATCH * EMB_DIM * HW);
}

// -------------------------------------------------------------------------
extern "C" void kernel_launch(void* const* d_in, const int* in_sizes, int n_in,
                              void* d_out, int out_size, void* d_ws, size_t ws_size,
                              hipStream_t stream) {
    const float* z      = (const float*)d_in[0];   // (64,64,32,32)
    const float* emb    = (const float*)d_in[1];   // (1024,64)
    const float* csz    = (const float*)d_in[2];   // (1024,)
    const float* ema_w  = (const float*)d_in[3];   // (1024,64)
    float* out = (float*)d_out;
    float* ws  = (float*)d_ws;

    float* enorm    = ws + WS_ENORM;
    float* counts   = ws + WS_COUNTS;
    float* dw       = ws + WS_DW;
    float* loss_sum = ws + WS_LOSS;

    // 1) zero the atomic accumulators (counts + dw + loss_sum contiguous)
    vq_zero_kernel<<<(WS_ZERON + 255) / 256, 256, 0, stream>>>(counts, WS_ZERON);

    // 2) codebook norms
    vq_enorm_kernel<<<NUM_EMB / 256, 256, 0, stream>>>(emb, enorm);

    // 3) WMMA distance GEMM + argmin  (dynamic LDS: full codebook, pitch 68)
    vq_argmin_kernel<<<NPTS / 256, 256, NUM_EMB * PITCH * sizeof(float), stream>>>(
        z, emb, enorm, out + OI);

    // 4) gather z_q, dw/counts segment sums, loss partials
    vq_scatter_kernel<<<(NPTS * EMB_DIM) / 256, 256, 0, stream>>>(
        z, emb, out + OI, out + OQ, counts, dw, loss_sum);

    // 5) EMA finalize
    vq_finalize_kernel<<<1, 1024, 0, stream>>>(
        csz, ema_w, counts, dw, loss_sum,
        out + OC, out + OW, out + OE, out + OL);
}